// GPNModel_37349035606695
// MI455X (gfx1250) — compile-verified
//
#include <hip/hip_runtime.h>
#include <math.h>

// ---- problem constants (match reference) ----
#define F_IN   256
#define HDIM   64
#define DLAT   16
#define NCLS   8
#define LLAY   10
#define KPROP  10
#define ALPHA_TP    0.1f
#define BETA_PRIOR  0.001f
#define BN_EPS      1e-5f
// -0.5 * D * ln(2*pi) with D=16
#define LOGP_CONST  (-14.7030163f)
// 0.5 * D * ln(4*pi) with D=16
#define LOG_SCALE_C (20.2481937f)

typedef __attribute__((ext_vector_type(2))) float v2f;
typedef __attribute__((ext_vector_type(8))) float v8f;
typedef __attribute__((ext_vector_type(4))) int   v4i_t;

// ---- CDNA5 async global->LDS staging support (guarded) ----
#if defined(__has_builtin)
#  if __has_builtin(__builtin_amdgcn_global_load_async_to_lds_b128)
#    define USE_ASYNC_STAGE 1
#  endif
#endif
#ifndef USE_ASYNC_STAGE
#  define USE_ASYNC_STAGE 0
#endif

__device__ __forceinline__ void wait_asynccnt0() {
#if defined(__has_builtin)
#  if __has_builtin(__builtin_amdgcn_s_wait_asynccnt)
  __builtin_amdgcn_s_wait_asynccnt(0);
#  else
  asm volatile("s_wait_asynccnt 0" ::: "memory");
#  endif
#else
  asm volatile("s_wait_asynccnt 0" ::: "memory");
#endif
}

__device__ __forceinline__ float softplusf(float x) {
  return (x > 20.f) ? x : log1pf(expf(x));
}

// =====================================================================
// GEMM1: h1_raw = x[N,256] @ W1[256,64] + b1   (fp32 WMMA 16x16x4)
// block = 128 threads (4 waves). Each block: 16 rows; wave w -> cols [16w,16w+16)
// x tile staged to LDS via async global->LDS copies (ASYNCcnt path).
// =====================================================================
__global__ void k_gemm1(const float* __restrict__ x, const float* __restrict__ W,
                        const float* __restrict__ bias, float* __restrict__ out) {
  __shared__ float xs[16 * F_IN];
  const int rowbase = blockIdx.x * 16;

#if USE_ASYNC_STAGE
  {
    const float* gsrc = x + (size_t)rowbase * F_IN;
    // 16*256 floats = 1024 x 16B chunks, 128 threads -> 8 chunks each
    for (int c = threadIdx.x; c < (16 * F_IN) / 4; c += 128) {
      const int i4 = c * 4;
      __builtin_amdgcn_global_load_async_to_lds_b128(
          (v4i_t*)(gsrc + i4), (v4i_t*)(xs + i4), /*offset=*/0, /*cpol=*/0);
    }
    wait_asynccnt0();
  }
#else
  for (int i = threadIdx.x; i < 16 * F_IN; i += blockDim.x)
    xs[i] = x[(size_t)rowbase * F_IN + i];
#endif
  __syncthreads();

  const int wave = threadIdx.x >> 5;
  const int lane = threadIdx.x & 31;
  const int g    = lane >> 4;          // lane half-group
  const int m    = lane & 15;          // A row for this lane
  const int col  = wave * 16 + (lane & 15);

  v8f acc;
  const float bv = bias[col];
#pragma unroll
  for (int r = 0; r < 8; ++r) acc[r] = bv;

#pragma unroll 8
  for (int k = 0; k < F_IN; k += 4) {
    const int kk = k + 2 * g;          // K index for VGPR j: kk + j
    v2f a; a.x = xs[m * F_IN + kk];     a.y = xs[m * F_IN + kk + 1];
    v2f b; b.x = W[kk * HDIM + col];    b.y = W[(kk + 1) * HDIM + col];
    acc = __builtin_amdgcn_wmma_f32_16x16x4_f32(false, a, false, b,
                                                (short)0, acc, false, false);
  }
#pragma unroll
  for (int r = 0; r < 8; ++r)
    out[(size_t)(rowbase + r + 8 * g) * HDIM + col] = acc[r];
}

// =====================================================================
// GEMM2: h2_raw = relu(bn(h1_raw)) @ W2[64,64] + b2
// BN fused into LDS staging: y = max(h*s + t, 0), st = [s[64], t[64]]
// =====================================================================
__global__ void k_gemm2(const float* __restrict__ hin, const float* __restrict__ st,
                        const float* __restrict__ W, const float* __restrict__ bias,
                        float* __restrict__ out) {
  __shared__ float hs[16 * HDIM];
  const int rowbase = blockIdx.x * 16;
  for (int i = threadIdx.x; i < 16 * HDIM; i += blockDim.x) {
    const int c = i & (HDIM - 1);
    const float v = hin[(size_t)rowbase * HDIM + i];
    hs[i] = fmaxf(fmaf(v, st[c], st[HDIM + c]), 0.f);
  }
  __syncthreads();

  const int wave = threadIdx.x >> 5;
  const int lane = threadIdx.x & 31;
  const int g    = lane >> 4;
  const int m    = lane & 15;
  const int col  = wave * 16 + (lane & 15);

  v8f acc;
  const float bv = bias[col];
#pragma unroll
  for (int r = 0; r < 8; ++r) acc[r] = bv;

#pragma unroll
  for (int k = 0; k < HDIM; k += 4) {
    const int kk = k + 2 * g;
    v2f a; a.x = hs[m * HDIM + kk];     a.y = hs[m * HDIM + kk + 1];
    v2f b; b.x = W[kk * HDIM + col];    b.y = W[(kk + 1) * HDIM + col];
    acc = __builtin_amdgcn_wmma_f32_16x16x4_f32(false, a, false, b,
                                                (short)0, acc, false, false);
  }
#pragma unroll
  for (int r = 0; r < 8; ++r)
    out[(size_t)(rowbase + r + 8 * g) * HDIM + col] = acc[r];
}

// =====================================================================
// GEMM3: z = relu(bn(h2_raw)) @ W3[64,16] + b3 -> z[N,16]
// block = 128 threads = 4 waves, each wave owns a 16-row tile (block: 64 rows)
// =====================================================================
__global__ void k_gemm3(const float* __restrict__ hin, const float* __restrict__ st,
                        const float* __restrict__ W, const float* __restrict__ bias,
                        float* __restrict__ zout, int n) {
  __shared__ float hs[64 * HDIM];
  const int blockrow = blockIdx.x * 64;
  for (int i = threadIdx.x; i < 64 * HDIM; i += blockDim.x) {
    const int rr = i >> 6;             // /HDIM
    const int c  = i & (HDIM - 1);
    const float v = (blockrow + rr < n) ? hin[(size_t)blockrow * HDIM + i] : 0.f;
    hs[i] = fmaxf(fmaf(v, st[c], st[HDIM + c]), 0.f);
  }
  __syncthreads();

  const int wave = threadIdx.x >> 5;
  const int lane = threadIdx.x & 31;
  const int rowbase = blockrow + wave * 16;
  if (rowbase < n) {                   // wave-uniform: EXEC stays all-ones
    const int g   = lane >> 4;
    const int m   = lane & 15;
    const int col = lane & 15;
    const float* A = hs + wave * 16 * HDIM;

    v8f acc;
    const float bv = bias[col];
#pragma unroll
    for (int r = 0; r < 8; ++r) acc[r] = bv;

#pragma unroll
    for (int k = 0; k < HDIM; k += 4) {
      const int kk = k + 2 * g;
      v2f a; a.x = A[m * HDIM + kk];     a.y = A[m * HDIM + kk + 1];
      v2f b; b.x = W[kk * DLAT + col];   b.y = W[(kk + 1) * DLAT + col];
      acc = __builtin_amdgcn_wmma_f32_16x16x4_f32(false, a, false, b,
                                                  (short)0, acc, false, false);
    }
#pragma unroll
    for (int r = 0; r < 8; ++r)
      zout[(size_t)(rowbase + r + 8 * g) * DLAT + col] = acc[r];
  }
}

// =====================================================================
// BatchNorm stats (biased variance, like torch BN training forward)
// =====================================================================
__global__ void k_zero(float* p, int cnt) {
  const int i = blockIdx.x * blockDim.x + threadIdx.x;
  if (i < cnt) p[i] = 0.f;
}

// gridDim.x*blockDim.x must be a multiple of 64 -> each thread has fixed column
__global__ void k_bn_accum(const float* __restrict__ h, float* __restrict__ sums, int n) {
  const int gsz = gridDim.x * blockDim.x;
  const int gid = blockIdx.x * blockDim.x + threadIdx.x;
  const int c   = gid & (HDIM - 1);
  float s = 0.f, ss = 0.f;
  const int total = n * HDIM;
  for (int i = gid; i < total; i += gsz) {
    const float v = h[i];
    s += v;
    ss = fmaf(v, v, ss);
  }
  atomicAdd(&sums[c], s);
  atomicAdd(&sums[HDIM + c], ss);
}

__global__ void k_bn_fin(const float* __restrict__ sums, const float* __restrict__ g,
                         const float* __restrict__ be, float* __restrict__ st, int n) {
  const int c = threadIdx.x;
  if (c < HDIM) {
    const float inv_n = 1.f / (float)n;
    const float mean = sums[c] * inv_n;
    const float var  = sums[HDIM + c] * inv_n - mean * mean;
    const float s = g[c] / sqrtf(var + BN_EPS);
    st[c]        = s;
    st[HDIM + c] = be[c] - mean * s;
  }
}

// =====================================================================
// Radial flow log-prob: one thread per (node, class)
// =====================================================================
__global__ void k_flow(const float* __restrict__ z, const float* __restrict__ x0,
                       const float* __restrict__ ap, const float* __restrict__ bp,
                       const float* __restrict__ mu, const float* __restrict__ lv,
                       float* __restrict__ logq, float* __restrict__ beta,
                       float* __restrict__ h0, int n) {
  const int idx = blockIdx.x * blockDim.x + threadIdx.x;
  if (idx >= n * NCLS) return;
  const int c  = idx & (NCLS - 1);
  const int nd = idx >> 3;

  float zz[DLAT];
#pragma unroll
  for (int d = 0; d < DLAT; ++d) zz[d] = z[(size_t)nd * DLAT + d];

  float log_det = 0.f;
  for (int l = LLAY - 1; l >= 0; --l) {
    const float a = softplusf(ap[l * NCLS + c]);
    const float b = softplusf(bp[l * NCLS + c]) - a;
    const float* x0l = x0 + (size_t)(l * NCLS + c) * DLAT;
    float diff[DLAT];
    float r2 = 0.f;
#pragma unroll
    for (int d = 0; d < DLAT; ++d) {
      diff[d] = zz[d] - x0l[d];
      r2 = fmaf(diff[d], diff[d], r2);
    }
    const float r  = fmaxf(sqrtf(r2), 1e-8f);
    const float hh = 1.f / (a + r);
    const float bh = b * hh;
#pragma unroll
    for (int d = 0; d < DLAT; ++d) zz[d] = fmaf(bh, diff[d], zz[d]);
    // h' = -1/(a+r)^2 = -hh^2 ; log_det += (D-1)*log1p(b*h) + log1p(b*h + b*h'*r)
    log_det += (float)(DLAT - 1) * log1pf(bh) + log1pf(bh - b * hh * hh * r);
  }

  float slv = 0.f, q = 0.f;
#pragma unroll
  for (int d = 0; d < DLAT; ++d) {
    const float l_ = lv[c * DLAT + d];
    slv += l_;
    const float dm = zz[d] - mu[c * DLAT + d];
    q = fmaf(dm * dm, expf(-l_), q);
  }
  const float lq = LOGP_CONST - 0.5f * slv - 0.5f * q + log_det;
  logq[idx] = lq;
  const float bf = expf(fminf(fmaxf(lq + LOG_SCALE_C, -30.f), 30.f));
  beta[idx] = bf;
  h0[idx]   = bf;
}

// =====================================================================
// APPNP: deg / symmetric norm / propagate with atomics
// =====================================================================
__global__ void k_deg_init(float* deg, int n) {
  const int i = blockIdx.x * blockDim.x + threadIdx.x;
  if (i < n) deg[i] = 1.f;   // self-loop contribution
}

__global__ void k_deg_acc(const int* __restrict__ row, float* __restrict__ deg, int e) {
  const int i = blockIdx.x * blockDim.x + threadIdx.x;
  if (i < e) atomicAdd(&deg[row[i]], 1.f);
}

__global__ void k_dis(float* deg, int n) {
  const int i = blockIdx.x * blockDim.x + threadIdx.x;
  if (i < n) {
    const float d = deg[i];
    deg[i] = (d > 0.f) ? rsqrtf(d) : 0.f;
  }
}

// msg[n][:] = dis[n]^2 * h[n][:]   (self-loop edge)
__global__ void k_msg_init(const float* __restrict__ dis, const float* __restrict__ h,
                           float* __restrict__ msg, int n) {
  const int i = blockIdx.x * blockDim.x + threadIdx.x;
  if (i < n) {
    const float w = dis[i] * dis[i];
    const float4 a = *(const float4*)(h + (size_t)i * NCLS);
    const float4 b = *(const float4*)(h + (size_t)i * NCLS + 4);
    float4 ma = make_float4(w * a.x, w * a.y, w * a.z, w * a.w);
    float4 mb = make_float4(w * b.x, w * b.y, w * b.z, w * b.w);
    *(float4*)(msg + (size_t)i * NCLS)     = ma;
    *(float4*)(msg + (size_t)i * NCLS + 4) = mb;
  }
}

__global__ void k_scatter(const int* __restrict__ row, const int* __restrict__ col,
                          const float* __restrict__ dis, const float* __restrict__ h,
                          float* __restrict__ msg, int e) {
  const int i = blockIdx.x * blockDim.x + threadIdx.x;
  if (i < e) {
    const int r  = row[i];
    const int cl = col[i];
    const float w = dis[r] * dis[cl];
    const float4 a = *(const float4*)(h + (size_t)cl * NCLS);
    const float4 b = *(const float4*)(h + (size_t)cl * NCLS + 4);
    float* mp = msg + (size_t)r * NCLS;
    atomicAdd(mp + 0, w * a.x);
    atomicAdd(mp + 1, w * a.y);
    atomicAdd(mp + 2, w * a.z);
    atomicAdd(mp + 3, w * a.w);
    atomicAdd(mp + 4, w * b.x);
    atomicAdd(mp + 5, w * b.y);
    atomicAdd(mp + 6, w * b.z);
    atomicAdd(mp + 7, w * b.w);
  }
}

__global__ void k_combine(const float* __restrict__ msg, const float* __restrict__ beta,
                          float* __restrict__ h, int cnt) {
  const int i = blockIdx.x * blockDim.x + threadIdx.x;
  if (i < cnt) h[i] = fmaf(1.f - ALPHA_TP, msg[i], ALPHA_TP * beta[i]);
}

__global__ void k_final(const float* __restrict__ h, float* __restrict__ alpha,
                        float* __restrict__ probs, int n) {
  const int i = blockIdx.x * blockDim.x + threadIdx.x;
  if (i < n) {
    float a[NCLS];
    float s = 0.f;
#pragma unroll
    for (int c = 0; c < NCLS; ++c) {
      a[c] = BETA_PRIOR + fmaxf(h[(size_t)i * NCLS + c], 0.f);
      s += a[c];
    }
    const float inv = 1.f / s;
#pragma unroll
    for (int c = 0; c < NCLS; ++c) {
      alpha[(size_t)i * NCLS + c] = a[c];
      probs[(size_t)i * NCLS + c] = a[c] * inv;
    }
  }
}

// =====================================================================
// host-side launch
// =====================================================================
extern "C" void kernel_launch(void* const* d_in, const int* in_sizes, int n_in,
                              void* d_out, int out_size, void* d_ws, size_t ws_size,
                              hipStream_t stream) {
  const float* x   = (const float*)d_in[0];
  const int*   ei  = (const int*)d_in[1];
  const float* W1  = (const float*)d_in[2];
  const float* b1  = (const float*)d_in[3];
  const float* g1  = (const float*)d_in[4];
  const float* be1 = (const float*)d_in[5];
  const float* W2  = (const float*)d_in[6];
  const float* b2  = (const float*)d_in[7];
  const float* g2  = (const float*)d_in[8];
  const float* be2 = (const float*)d_in[9];
  const float* W3  = (const float*)d_in[10];
  const float* b3  = (const float*)d_in[11];
  const float* x0  = (const float*)d_in[12];
  const float* ap  = (const float*)d_in[13];
  const float* bp  = (const float*)d_in[14];
  const float* mu  = (const float*)d_in[15];
  const float* lv  = (const float*)d_in[16];

  const int n = in_sizes[0] / F_IN;     // 100000
  const int e = in_sizes[1] / 2;        // 3200000
  const int* row = ei;
  const int* col = ei + e;

  // outputs: alpha[N,C], probs[N,C], z[N,D], beta_ft[N,C], log_q[N,C]
  float* out     = (float*)d_out;
  float* alpha_o = out;
  float* probs_o = alpha_o + (size_t)n * NCLS;
  float* z_o     = probs_o + (size_t)n * NCLS;
  float* beta_o  = z_o     + (size_t)n * DLAT;
  float* logq_o  = beta_o  + (size_t)n * NCLS;

  // workspace layout (floats), 16B-aligned sections
  float* ws   = (float*)d_ws;
  float* h1   = ws;                                  // n*64
  float* h2   = h1 + (size_t)n * HDIM;               // n*64
  float* sums = h2 + (size_t)n * HDIM;               // 128
  float* st1  = sums + 128;                          // 128
  float* st2  = st1 + 128;                           // 128
  float* dis  = st2 + 128;                           // n (rounded to 4)
  const int n4 = (n + 3) & ~3;
  float* hbuf = dis + n4;                            // n*8
  float* msg  = hbuf + (size_t)n * NCLS;             // n*8

  const int TPB = 256;
  const int gN  = (n + TPB - 1) / TPB;
  const int gNC = (n * NCLS + TPB - 1) / TPB;
  const int gE  = (e + TPB - 1) / TPB;

  // ---- encoder ----
  k_gemm1<<<n / 16, 128, 0, stream>>>(x, W1, b1, h1);
  k_zero<<<1, 128, 0, stream>>>(sums, 128);
  k_bn_accum<<<512, 256, 0, stream>>>(h1, sums, n);
  k_bn_fin<<<1, 64, 0, stream>>>(sums, g1, be1, st1, n);

  k_gemm2<<<n / 16, 128, 0, stream>>>(h1, st1, W2, b2, h2);
  k_zero<<<1, 128, 0, stream>>>(sums, 128);
  k_bn_accum<<<512, 256, 0, stream>>>(h2, sums, n);
  k_bn_fin<<<1, 64, 0, stream>>>(sums, g2, be2, st2, n);

  k_gemm3<<<(n + 63) / 64, 128, 0, stream>>>(h2, st2, W3, b3, z_o, n);

  // ---- flow density ----
  k_flow<<<gNC, TPB, 0, stream>>>(z_o, x0, ap, bp, mu, lv, logq_o, beta_o, hbuf, n);

  // ---- APPNP ----
  k_deg_init<<<gN, TPB, 0, stream>>>(dis, n);
  k_deg_acc<<<gE, TPB, 0, stream>>>(row, dis, e);
  k_dis<<<gN, TPB, 0, stream>>>(dis, n);
  for (int k = 0; k < KPROP; ++k) {
    k_msg_init<<<gN, TPB, 0, stream>>>(dis, hbuf, msg, n);
    k_scatter<<<gE, TPB, 0, stream>>>(row, col, dis, hbuf, msg, e);
    k_combine<<<gNC, TPB, 0, stream>>>(msg, beta_o, hbuf, n * NCLS);
  }

  k_final<<<gN, TPB, 0, stream>>>(hbuf, alpha_o, probs_o, n);
}